// AlphaModel_86114094285088
// MI455X (gfx1250) — compile-verified
//
#include <hip/hip_runtime.h>

typedef __attribute__((ext_vector_type(2))) float v2f;
typedef __attribute__((ext_vector_type(8))) float v8f;

#define GB   2048
#define GP   16384
#define FXY  2343.0242837919386f
#define CXc  1280.0f
#define CYc  720.0f

// ---------------------------------------------------------------------------
// Per-Gaussian coefficient vector G[b][8] such that
//   inside[b,p] = sum_k G[b][k] * F[k][p]
// with pixel features F = [u^2, v^2, u*v, u, v, 1, 0, 0], u = tx-cx, v = ty-cy.
// ---------------------------------------------------------------------------
__global__ void prep_G_kernel(const float* __restrict__ mh,
                              const float* __restrict__ xmat,
                              const float* __restrict__ cw,
                              float* __restrict__ G) {
  int b = blockIdx.x * blockDim.x + threadIdx.x;
  if (b >= GB) return;
  const float* xb = xmat + b * 16;
  float m0 = mh[0], m1 = mh[1], m2 = mh[2], m3 = mh[3];
  float r00=xb[0],  r01=xb[1],  r02=xb[2],  t0=xb[3];
  float r10=xb[4],  r11=xb[5],  r12=xb[6],  t1=xb[7];
  float r20=xb[8],  r21=xb[9],  r22=xb[10], t2=xb[11];
  // means_cam = x[:, :3, :] @ means_hom
  float xm = r00*m0 + r01*m1 + r02*m2 + t0*m3;
  float ym = r10*m0 + r11*m1 + r12*m2 + t1*m3;
  float zc = r20*m0 + r21*m1 + r22*m2 + t2*m3;
  float C00=cw[0],C01=cw[1],C02=cw[2];
  float C10=cw[3],C11=cw[4],C12=cw[5];
  float C20=cw[6],C21=cw[7],C22=cw[8];
  // RC = R * cov_world
  float rc00 = r00*C00 + r01*C10 + r02*C20;
  float rc01 = r00*C01 + r01*C11 + r02*C21;
  float rc02 = r00*C02 + r01*C12 + r02*C22;
  float rc10 = r10*C00 + r11*C10 + r12*C20;
  float rc11 = r10*C01 + r11*C11 + r12*C21;
  float rc12 = r10*C02 + r11*C12 + r12*C22;
  float rc20 = r20*C00 + r21*C10 + r22*C20;
  float rc21 = r20*C01 + r21*C11 + r22*C21;
  float rc22 = r20*C02 + r21*C12 + r22*C22;
  // cov_cam = RC * R^T (symmetric)
  float c00 = rc00*r00 + rc01*r01 + rc02*r02;
  float c01 = rc00*r10 + rc01*r11 + rc02*r12;
  float c02 = rc00*r20 + rc01*r21 + rc02*r22;
  float c11 = rc10*r10 + rc11*r11 + rc12*r12;
  float c12 = rc10*r20 + rc11*r21 + rc12*r22;
  float c22 = rc20*r20 + rc21*r21 + rc22*r22;
  float z2  = zc * zc;
  float J00 = zc*FXY, J02 = -FXY*xm, J11 = zc*FXY, J12 = -FXY*ym;
  float a2 = J00*J00*c00 + 2.f*J00*J02*c02 + J02*J02*c22;
  float d2 = J11*J11*c11 + 2.f*J11*J12*c12 + J12*J12*c22;
  float b2 = J00*J11*c01 + J00*J12*c02 + J02*J11*c12 + J02*J12*c22;
  float det  = a2*d2 - b2*b2;
  float co00 = d2/det, co01 = -b2/det, co11 = a2/det;
  // dx_x = z2*tx - z0*mp0 = z2*u + hx  with hx = z0*J02 (exact cancellation of cx terms)
  float hx = zc * J02;
  float hy = zc * J12;
  float z4 = z2 * z2;
  float* g = G + b*8;
  g[0] = co00 * z4;
  g[1] = co11 * z4;
  g[2] = 2.f * co01 * z4;
  g[3] = 2.f * z2 * (co00*hx + co01*hy);
  g[4] = 2.f * z2 * (co11*hy + co01*hx);
  g[5] = co00*hx*hx + co11*hy*hy + 2.f*co01*hx*hy;
  g[6] = 0.f;
  g[7] = 0.f;
}

// Pixel feature matrix, K-major: F[k*GP + p]
__global__ void prep_F_kernel(const float* __restrict__ tc,
                              float* __restrict__ F) {
  int p = blockIdx.x * blockDim.x + threadIdx.x;
  if (p >= GP) return;
  float u = tc[2*p]     - CXc;
  float v = tc[2*p + 1] - CYc;
  F[0*GP + p] = u*u;
  F[1*GP + p] = v*v;
  F[2*GP + p] = u*v;
  F[3*GP + p] = u;
  F[4*GP + p] = v;
  F[5*GP + p] = 1.f;
  F[6*GP + p] = 0.f;
  F[7*GP + p] = 0.f;
}

// ---------------------------------------------------------------------------
// Rank-8 GEMM: out(2048 x 16384) = G(2048 x 8) * F(8 x 16384)
// via V_WMMA_F32_16X16X4_F32 (full fp32 A/B, fp32 accumulate).
// Block = 256 threads = 8 waves; wave w handles a 16(M) x 128(N) strip,
// reusing its A fragment across 8 N-tiles. Grid = (16, 128).
// ---------------------------------------------------------------------------
__global__ void __launch_bounds__(256)
splat_wmma_kernel(const float* __restrict__ G,
                  const float* __restrict__ F,
                  float* __restrict__ out) {
  const int lane   = threadIdx.x & 31;
  const int wave   = threadIdx.x >> 5;
  const int m_base = blockIdx.y << 4;                    // 16 rows per tile
  const int n_base = (blockIdx.x << 10) + (wave << 7);   // 128 cols per wave
  const int khalf  = (lane >> 4) << 1;                   // 0 (lanes 0-15) or 2
  const int nlane  = lane & 15;

  // A fragment (16x4 f32): lane<16 -> K=0,1 ; lane>=16 -> K=2,3 (and +4 for 2nd op)
  const int row = m_base + nlane;
  v2f a0, a1;
  a0.x = G[row*8 + khalf];         a0.y = G[row*8 + khalf + 1];
  a1.x = G[row*8 + 4 + khalf];     a1.y = G[row*8 + 4 + khalf + 1];

  const int mrow = m_base + ((lane >> 4) << 3);          // D: M = v + 8*(lane>=16)

  #pragma unroll
  for (int t = 0; t < 8; ++t) {
    const int p = n_base + (t << 4) + nlane;
    // B fragment (4x16 f32): VGPR v holds K=v (lanes 0-15) / K=v+2 (lanes 16-31)
    v2f b0, b1;
    b0.x = F[(khalf + 0)*GP + p];  b0.y = F[(khalf + 1)*GP + p];
    b1.x = F[(khalf + 4)*GP + p];  b1.y = F[(khalf + 5)*GP + p];

    v8f c = {0.f, 0.f, 0.f, 0.f, 0.f, 0.f, 0.f, 0.f};
    c = __builtin_amdgcn_wmma_f32_16x16x4_f32(false, a0, false, b0,
                                              (short)0, c, false, false);
    c = __builtin_amdgcn_wmma_f32_16x16x4_f32(false, a1, false, b1,
                                              (short)0, c, false, false);
    #pragma unroll
    for (int v = 0; v < 8; ++v)
      out[(mrow + v) * GP + p] = c[v];
  }
}

extern "C" void kernel_launch(void* const* d_in, const int* in_sizes, int n_in,
                              void* d_out, int out_size, void* d_ws, size_t ws_size,
                              hipStream_t stream) {
  const float* mh = (const float*)d_in[0];   // means_hom_tmp (4)
  const float* x  = (const float*)d_in[1];   // x (2048*16)
  const float* cw = (const float*)d_in[2];   // cov_world (9)
  // d_in[3] = opacities_rast : unused by the reference output
  const float* tc = (const float*)d_in[4];   // tile_coord (16384*2)

  float* G = (float*)d_ws;                   // 2048*8 floats   (64 KiB)
  float* F = G + GB * 8;                     // 8*16384 floats  (512 KiB)
  float* out = (float*)d_out;                // 2048*16384 floats

  prep_G_kernel<<<GB / 256, 256, 0, stream>>>(mh, x, cw, G);
  prep_F_kernel<<<GP / 256, 256, 0, stream>>>(tc, F);
  splat_wmma_kernel<<<dim3(16, 128), 256, 0, stream>>>(G, F, out);
}